// HgnnEncoder_43353399886444
// MI455X (gfx1250) — compile-verified
//
#include <hip/hip_runtime.h>

typedef float v2f __attribute__((ext_vector_type(2)));
typedef float v8f __attribute__((ext_vector_type(8)));

#define N_NODES 50000
#define N_EDGES 10000

// ---------------- zero ----------------
__global__ void hgnn_zero4(float* __restrict__ p, int n4) {
    int i = blockIdx.x * blockDim.x + threadIdx.x;
    if (i < n4) ((float4*)p)[i] = make_float4(0.f, 0.f, 0.f, 0.f);
}

// ---------------- degrees ----------------
__global__ void hgnn_degrees(const int* __restrict__ node_idx,
                             const int* __restrict__ edge_idx,
                             float* __restrict__ deg_v, float* __restrict__ deg_e,
                             int nnz) {
    int i = blockIdx.x * blockDim.x + threadIdx.x;
    if (i < nnz) {
        atomicAdd(&deg_v[node_idx[i]], 1.0f);
        atomicAdd(&deg_e[edge_idx[i]], 1.0f);
    }
}

__global__ void hgnn_invert(float* __restrict__ d, int n) {
    int i = blockIdx.x * blockDim.x + threadIdx.x;
    if (i < n) {
        float v = d[i];
        d[i] = (v > 0.f) ? (1.0f / v) : 0.0f;
    }
}

// ---------------- GEMM via V_WMMA_F32_16X16X4_F32, W staged in LDS ----------
// out[M,FOUT] = A[M,FIN] @ W[FIN,FOUT] + bias[FOUT]
// One wave owns a 16-row strip and accumulates ALL FOUT/16 column tiles,
// so each A fragment (2 VGPRs) feeds FOUT/16 back-to-back WMMAs.
// B fragments come from LDS (W loaded cooperatively once per block).
template <int FIN, int FOUT>
__global__ void hgnn_gemm_wmma_lds(const float* __restrict__ A,
                                   const float* __restrict__ W,
                                   const float* __restrict__ bias,
                                   float* __restrict__ out,
                                   int mTiles) {
    constexpr int TN = FOUT / 16;           // column tiles per wave (<= 8)
    __shared__ float wlds[FIN * FOUT];      // <= 64 KB (WGP has 320 KB)

    // cooperative stage of the whole weight matrix into LDS
    for (int idx = threadIdx.x; idx < (FIN * FOUT) / 4; idx += blockDim.x)
        ((float4*)wlds)[idx] = ((const float4*)W)[idx];
    __syncthreads();

    int wave = threadIdx.x >> 5;
    int lane = threadIdx.x & 31;
    int tm   = blockIdx.x * (blockDim.x >> 5) + wave;
    if (tm >= mTiles) return;               // wave-uniform: EXEC all-ones for WMMA

    int hi   = lane >> 4;                   // 0: K pair {0,1}; 1: K pair {2,3}
    int l15  = lane & 15;
    int arow = tm * 16 + l15;

    v8f zero = {};
    v8f acc[TN];
#pragma unroll
    for (int t = 0; t < TN; ++t) acc[t] = zero;

    const float* arp = A + (long)arow * FIN;
#pragma unroll 4
    for (int k = 0; k < FIN; k += 4) {
        int ka = k + 2 * hi;
        v2f a = *(const v2f*)(arp + ka);    // global_load_b64, reused TN times
#pragma unroll
        for (int t = 0; t < TN; ++t) {
            int bcol = t * 16 + l15;
            v2f b;
            b.x = wlds[ka * FOUT + bcol];          // ds_load_b32
            b.y = wlds[(ka + 1) * FOUT + bcol];
            acc[t] = __builtin_amdgcn_wmma_f32_16x16x4_f32(
                false, a, false, b, (short)0, acc[t], false, false);
        }
    }

#pragma unroll
    for (int t = 0; t < TN; ++t) {
        float bv = bias[t * 16 + l15];
#pragma unroll
        for (int r = 0; r < 8; ++r) {       // C/D layout: VGPR r -> row r (lanes 0-15) / r+8
            int orow = tm * 16 + r + 8 * hi;
            out[(long)orow * FOUT + t * 16 + l15] = acc[t][r] + bv;
        }
    }
}

// ---------------- gather + atomic scatter-add ----------------
// dst[to[i], :] += src[from[i], :] * (scale ? scale[from[i]] : 1)
// one thread handles 4 consecutive features of one incidence;
// a wave covers one (F=128) incidence row -> coalesced b128 gather, L2 atomics
__global__ void hgnn_scatter_add(const float* __restrict__ src,
                                 const int* __restrict__ from_idx,
                                 const int* __restrict__ to_idx,
                                 const float* __restrict__ scale,
                                 float* __restrict__ dst,
                                 int F, int nnz) {
    int chunks = F >> 2;
    int gid = blockIdx.x * blockDim.x + threadIdx.x;
    int total = nnz * chunks;
    if (gid >= total) return;
    int i = gid / chunks;
    int c = gid - i * chunks;
    int from = from_idx[i];
    int to   = to_idx[i];
    float4 v = *(const float4*)(src + (long)from * F + c * 4);
    float s = scale ? scale[from] : 1.0f;
    float* d = dst + (long)to * F + c * 4;
    atomicAdd(d + 0, v.x * s);
    atomicAdd(d + 1, v.y * s);
    atomicAdd(d + 2, v.z * s);
    atomicAdd(d + 3, v.w * s);
}

// ---------------- finalize: buf = relu(buf * Dinv[row]) ----------------
__global__ void hgnn_finalize(float* __restrict__ buf, const float* __restrict__ dinv,
                              int F, int n) {
    int i = blockIdx.x * blockDim.x + threadIdx.x;
    if (i < n) {
        int row = i / F;
        float v = buf[i] * dinv[row];
        buf[i] = v > 0.f ? v : 0.f;
    }
}

static inline int cdiv(int a, int b) { return (a + b - 1) / b; }

extern "C" void kernel_launch(void* const* d_in, const int* in_sizes, int n_in,
                              void* d_out, int out_size, void* d_ws, size_t ws_size,
                              hipStream_t stream) {
    const float* x  = (const float*)d_in[0];
    const int*   eg = (const int*)d_in[1];
    const float* w1 = (const float*)d_in[2];
    const float* b1 = (const float*)d_in[3];
    const float* w2 = (const float*)d_in[4];
    const float* b2 = (const float*)d_in[5];
    const float* w3 = (const float*)d_in[6];
    const float* b3 = (const float*)d_in[7];

    const int nnz = in_sizes[1] / 2;
    const int* node_idx = eg;            // edge[0] = node indices
    const int* edge_idx = eg + nnz;      // edge[1] = hyperedge indices

    // workspace carve (all offsets multiples of 4 floats -> 16B aligned)
    float* f = (float*)d_ws;
    float* deg_e = f; f += N_EDGES;            // -> Binv after invert
    float* deg_v = f; f += N_NODES;            // -> Dinv after invert
    float* me    = f; f += (long)N_EDGES * 128;
    float* bufA  = f; f += (long)N_NODES * 128;
    float* bufB  = f; f += (long)N_NODES * 128;

    const int TB = 256;
    const int mTiles = N_NODES / 16;           // 3125 exactly

    // degrees -> inverse degrees (constant across layers)
    hgnn_zero4<<<cdiv(N_EDGES / 4, TB), TB, 0, stream>>>(deg_e, N_EDGES / 4);
    hgnn_zero4<<<cdiv(N_NODES / 4, TB), TB, 0, stream>>>(deg_v, N_NODES / 4);
    hgnn_degrees<<<cdiv(nnz, TB), TB, 0, stream>>>(node_idx, edge_idx, deg_v, deg_e, nnz);
    hgnn_invert<<<cdiv(N_EDGES, TB), TB, 0, stream>>>(deg_e, N_EDGES);
    hgnn_invert<<<cdiv(N_NODES, TB), TB, 0, stream>>>(deg_v, N_NODES);

    // one hypergraph-conv layer (post-GEMM buffer = bufA)
    auto post_gemm = [&](int Fout, float* outbuf) {
        int meN = N_EDGES * Fout;
        hgnn_zero4<<<cdiv(meN / 4, TB), TB, 0, stream>>>(me, meN / 4);
        hgnn_scatter_add<<<cdiv(nnz * (Fout / 4), TB), TB, 0, stream>>>(
            bufA, node_idx, edge_idx, (const float*)nullptr, me, Fout, nnz);

        int outN = N_NODES * Fout;
        hgnn_zero4<<<cdiv(outN / 4, TB), TB, 0, stream>>>(outbuf, outN / 4);
        hgnn_scatter_add<<<cdiv(nnz * (Fout / 4), TB), TB, 0, stream>>>(
            me, edge_idx, node_idx, deg_e /*Binv*/, outbuf, Fout, nnz);

        hgnn_finalize<<<cdiv(outN, TB), TB, 0, stream>>>(outbuf, deg_v /*Dinv*/, Fout, outN);
    };

    const int gemmBlocks = cdiv(mTiles, TB / 32);   // 8 waves per block

    // layer 1: x[50000,128] -> bufB[50000,128]
    hgnn_gemm_wmma_lds<128, 128><<<gemmBlocks, TB, 0, stream>>>(x, w1, b1, bufA, mTiles);
    post_gemm(128, bufB);

    // layer 2: bufB[.,128] -> bufB[.,64]  (gemm reads bufB before scatter overwrites it)
    hgnn_gemm_wmma_lds<128, 64><<<gemmBlocks, TB, 0, stream>>>(bufB, w2, b2, bufA, mTiles);
    post_gemm(64, bufB);

    // layer 3: bufB[.,64] -> d_out[50000,32]
    hgnn_gemm_wmma_lds<64, 32><<<gemmBlocks, TB, 0, stream>>>(bufB, w3, b3, bufA, mTiles);
    post_gemm(32, (float*)d_out);
}